// GCNLayer_62096637165579
// MI455X (gfx1250) — compile-verified
//
#include <hip/hip_runtime.h>
#include <hip/hip_bf16.h>

#define TILE  1024
#define BLOCK 256

#define AS1 __attribute__((address_space(1)))
#define AS3 __attribute__((address_space(3)))

typedef int v2i __attribute__((vector_size(8)));    // builtin b64 param: "int __vector(2)"
typedef int v4i __attribute__((vector_size(16)));   // builtin b128 param: "int __vector(4)"

// ---------------------------------------------------------------------------
// Never launched. Placed FIRST in the module purely so the disasm snippet
// reveals how the relaxed agent-scope f32 atomic add lowers on gfx1250
// (expect a single non-returning global_atomic_add_f32, not a CAS loop).
// ---------------------------------------------------------------------------
__global__ void probe_atomic_fadd(float* p, float v) {
    (void)__hip_atomic_fetch_add(p + threadIdx.x, v, __ATOMIC_RELAXED, __HIP_MEMORY_SCOPE_AGENT);
}

// Generic->AS1: global generic addresses are numerically identical to AS1.
// Generic->AS3: for LDS, generic addr low 32 bits == LDS byte offset (aperture rule).
__device__ __forceinline__ void async_copy_b128(const void* g, void* l) {
#if __has_builtin(__builtin_amdgcn_global_load_async_to_lds_b128)
    __builtin_amdgcn_global_load_async_to_lds_b128(
        (AS1 v4i*)(unsigned long long)g,
        (AS3 v4i*)(unsigned int)(unsigned long long)l, 0, 0);
#endif
}
__device__ __forceinline__ void async_copy_b32(const void* g, void* l) {
#if __has_builtin(__builtin_amdgcn_global_load_async_to_lds_b32)
    __builtin_amdgcn_global_load_async_to_lds_b32(
        (AS1 int*)(unsigned long long)g,
        (AS3 int*)(unsigned int)(unsigned long long)l, 0, 0);
#endif
}

__device__ __forceinline__ void wait_async_zero() {
#if __has_builtin(__builtin_amdgcn_s_wait_asynccnt)
    __builtin_amdgcn_s_wait_asynccnt(0);
#else
    asm volatile("s_wait_asynccnt 0" ::: "memory");
#endif
}

// ---------------------------------------------------------------------------
// Edge-parallel COO SpMM scatter:
//   out[rows[e]] += vals[e] * embeds[cols[e]]   (E = 1.6M, D = 48)
// One 1024-edge tile per 256-thread block; tile staged via async global->LDS
// (ASYNCcnt path); 16 lanes per edge x 3 features per lane (48 = 3*16) gives
// 64B-coalesced L2-resident gathers and coalesced f32 atomics.
// ---------------------------------------------------------------------------
__global__ __launch_bounds__(BLOCK) void gcn_spmm_scatter_kernel(
    const int* __restrict__ rows, const int* __restrict__ cols,
    const float* __restrict__ vals, const float* __restrict__ embeds,
    float* __restrict__ out, int nEdges)
{
    __shared__ int   s_rows[TILE];
    __shared__ int   s_cols[TILE];
    __shared__ float s_vals[TILE];

    const int tid      = threadIdx.x;
    const int tileBase = blockIdx.x * TILE;

    // ---- Stage this block's 1024-edge tile into LDS (3 x B128 per thread) ----
#if __has_builtin(__builtin_amdgcn_global_load_async_to_lds_b128)
    {
        const int k = tid * 4;              // 0,4,...,1020 -> covers TILE with 256 threads
        const int e = tileBase + k;
        if (e + 3 < nEdges) {
            async_copy_b128(rows + e, &s_rows[k]);
            async_copy_b128(cols + e, &s_cols[k]);
            async_copy_b128(vals + e, &s_vals[k]);
        } else {
            // tail: per-element B32 async copies for the straddling quad
            for (int j = 0; j < 4; ++j) {
                if (e + j < nEdges) {
                    async_copy_b32(rows + e + j, &s_rows[k + j]);
                    async_copy_b32(cols + e + j, &s_cols[k + j]);
                    async_copy_b32(vals + e + j, &s_vals[k + j]);
                }
            }
        }
        wait_async_zero();                  // this wave's async writes have landed in LDS
    }
#else
    for (int k = tid; k < TILE; k += BLOCK) {
        int e = tileBase + k;
        if (e < nEdges) {
            s_rows[k] = rows[e];
            s_cols[k] = cols[e];
            s_vals[k] = vals[e];
        }
    }
#endif
    __syncthreads();

    // ---- Scatter: 16 lanes per edge, 3 features per lane (48 = 3*16) ----
    const int lane = tid & 31;
    const int wave = tid >> 5;              // 0..7 (wave32)
    const int half = lane >> 4;             // 0/1: which edge of the wave's pair
    const int f    = lane & 15;             // feature sub-index
    const int slot = wave * 2 + half;       // 0..15

    for (int i = 0; i < TILE; i += 16) {
        const int k = i + slot;
        const int e = tileBase + k;
        if (e < nEdges) {
            const int   r = s_rows[k];      // LDS broadcast within half-wave
            const int   c = s_cols[k];
            const float v = s_vals[k];
            const float* __restrict__ src = embeds + (size_t)c * 48 + f;  // 64B coalesced / half-wave
            float*       __restrict__ dst = out    + (size_t)r * 48 + f;
            const float a0 = src[0]  * v;
            const float a1 = src[16] * v;
            const float a2 = src[32] * v;
            // relaxed agent-scope FP32 atomics -> non-returning global_atomic_add_f32 (L2 atomic units)
            (void)__hip_atomic_fetch_add(dst,      a0, __ATOMIC_RELAXED, __HIP_MEMORY_SCOPE_AGENT);
            (void)__hip_atomic_fetch_add(dst + 16, a1, __ATOMIC_RELAXED, __HIP_MEMORY_SCOPE_AGENT);
            (void)__hip_atomic_fetch_add(dst + 32, a2, __ATOMIC_RELAXED, __HIP_MEMORY_SCOPE_AGENT);
        }
    }
}

__global__ __launch_bounds__(BLOCK) void gcn_zero_kernel(float* __restrict__ out, int n) {
    int n4 = n >> 2;
    float4* o4 = (float4*)out;
    int stride = gridDim.x * blockDim.x;
    for (int i = blockIdx.x * blockDim.x + threadIdx.x; i < n4; i += stride)
        o4[i] = make_float4(0.f, 0.f, 0.f, 0.f);
    // scalar tail (out_size not multiple of 4 — not the case here, but be safe)
    if (blockIdx.x == 0 && threadIdx.x == 0)
        for (int i = n4 << 2; i < n; ++i) out[i] = 0.f;
}

extern "C" void kernel_launch(void* const* d_in, const int* in_sizes, int n_in,
                              void* d_out, int out_size, void* d_ws, size_t ws_size,
                              hipStream_t stream) {
    const int*   rows   = (const int*)d_in[0];
    const int*   cols   = (const int*)d_in[1];
    const float* vals   = (const float*)d_in[2];
    const float* embeds = (const float*)d_in[3];
    float*       out    = (float*)d_out;

    const int nEdges = in_sizes[0];

    // 1) zero the accumulator (atomic scatter accumulates into it)
    int n4 = out_size >> 2;
    int zblocks = (n4 + BLOCK - 1) / BLOCK;
    if (zblocks < 1) zblocks = 1;
    gcn_zero_kernel<<<zblocks, BLOCK, 0, stream>>>(out, out_size);

    // 2) edge-parallel scatter, one 1024-edge tile per block
    int nTiles = (nEdges + TILE - 1) / TILE;
    gcn_spmm_scatter_kernel<<<nTiles, BLOCK, 0, stream>>>(rows, cols, vals, embeds, out, nEdges);
}